// TimeSeriesAugmenter_15908558865315
// MI455X (gfx1250) — compile-verified
//
#include <hip/hip_runtime.h>

// Problem constants (match reference)
constexpr int B         = 64;
constexpr int T         = 4096;
constexpr int C         = 128;
constexpr int MAX_SHIFT = 409;          // int(T * 0.1)
constexpr float NOISE_STD = 0.1f;

constexpr int C4          = C / 4;      // 32 float4 per time step == one wave32 row
constexpr int T_PER_BLOCK = 64;         // time steps per block
constexpr int THREADS     = 256;        // 8 wave32s; each wave covers 8 time steps

// Native clang vector so __builtin_nontemporal_{load,store} accepts it and it
// lowers to a single global_{load,store}_b128 with th:TH_*_NT.
typedef __attribute__((ext_vector_type(4))) float f4;

__global__ __launch_bounds__(THREADS)
void TimeSeriesAugmenter_15908558865315_kernel(const float* __restrict__ x,
                                               const float* __restrict__ noise,
                                               const float* __restrict__ scale,
                                               const int*   __restrict__ shifts,
                                               float*       __restrict__ out)
{
    constexpr int blocksPerBatch = T / T_PER_BLOCK;      // 64
    const int b  = blockIdx.x / blocksPerBatch;          // batch sample (block-uniform)
    const int t0 = (blockIdx.x % blocksPerBatch) * T_PER_BLOCK;

    // Block-uniform params -> scalar loads on gfx1250.
    const float sc = scale[b];
    const int   s  = shifts[b] - MAX_SHIFT;              // in [-409, 409]

    // Wave index is uniform across a wave32; readfirstlane forces the compiler
    // to keep all time-step / wrap arithmetic in SALU and use scalar-base
    // addressing for the VMEM ops. Only the lane offset stays vector.
    const int wave = __builtin_amdgcn_readfirstlane((int)(threadIdx.x >> 5)); // 0..7
    const int lane = threadIdx.x & 31;                   // channel-quad index (c4)

    const size_t base = (size_t)b * T * C;
    const f4* __restrict__ x4 = (const f4*)(x     + base);
    const f4* __restrict__ n4 = (const f4*)(noise + base);
    f4*       __restrict__ o4 = (f4*)      (out   + base);

    // Each wave handles 8 time steps; src_t is wave-uniform (SGPR), lanes
    // stride the 128-channel row as 32 contiguous float4s (512B per access).
#pragma unroll
    for (int i = 0; i < T_PER_BLOCK / 8; ++i) {
        const int t = t0 + i * 8 + wave;                 // scalar
        int src_t = t - s;                               // |s| < T -> at most one wrap
        if (src_t < 0)  src_t += T;                      // s_cmp/s_cselect
        if (src_t >= T) src_t -= T;

        const f4 xv = __builtin_nontemporal_load(&x4[(size_t)src_t * C4 + lane]);
        const f4 nv = __builtin_nontemporal_load(&n4[(size_t)src_t * C4 + lane]);

        const f4 r = (xv + nv * NOISE_STD) * sc;         // 3 flops/elem, BW-bound

        __builtin_nontemporal_store(r, &o4[(size_t)t * C4 + lane]);
    }
}

extern "C" void kernel_launch(void* const* d_in, const int* in_sizes, int n_in,
                              void* d_out, int out_size, void* d_ws, size_t ws_size,
                              hipStream_t stream)
{
    (void)in_sizes; (void)n_in; (void)out_size; (void)d_ws; (void)ws_size;

    const float* x      = (const float*)d_in[0];   // [B,T,C] f32
    const float* noise  = (const float*)d_in[1];   // [B,T,C] f32
    const float* scale  = (const float*)d_in[2];   // [B,1,1] f32
    const int*   shifts = (const int*)  d_in[3];   // [B] i32
    float*       out    = (float*)d_out;           // [B,T,C] f32

    const int grid = B * (T / T_PER_BLOCK);        // 4096 blocks, exact cover
    TimeSeriesAugmenter_15908558865315_kernel<<<grid, THREADS, 0, stream>>>(
        x, noise, scale, shifts, out);
}